// RNN_14534169329796
// MI455X (gfx1250) — compile-verified
//
#include <hip/hip_runtime.h>
#include <hip/hip_bf16.h>

// ---------------------------------------------------------------------------
// Problem constants (from the reference)
// ---------------------------------------------------------------------------
#define T_LEN 131072
#define IN_F  121
#define KPAD  124          // K padded to multiple of 4
#define HID   20
#define G3    (3 * HID)    // 60 gate rows (r, z, n)
#define NLAY  4

typedef __attribute__((ext_vector_type(2))) float v2f;
typedef __attribute__((ext_vector_type(8))) float v8f;

// ---------------------------------------------------------------------------
// Kernel 1: gxT = (x @ w_ih0^T + b_ih0)^T   via V_WMMA_F32_16X16X4_F32
//   x: [T, 121]   w: [60, 121]   bias: [60]   gxT: [60, T]  (transposed!)
// Both operands staged in zero-padded LDS so the K-loop is branch-free:
// 31 x (2 ds_load_b64 + v_wmma). One wave = one 16x16 tile; 4 waves cover
// N=60 (padded to 64); blockIdx.x = M-tile.
// ---------------------------------------------------------------------------
__global__ __launch_bounds__(128) void gru_in_gemm(
    const float* __restrict__ X,
    const float* __restrict__ W,
    const float* __restrict__ bias,
    float* __restrict__ gxT)
{
    __shared__ float Xl[16][KPAD];     //  7.9 KB
    __shared__ float Wl[64][KPAD];     // 31.7 KB (rows 60..63 stay zero)

    const int tid   = threadIdx.x;
    const int lane  = tid & 31;
    const int wave  = tid >> 5;        // N-tile 0..3
    const int l16   = lane & 15;
    const int khalf = lane >> 4;       // 0: K={k0,k0+1}, 1: K={k0+2,k0+3}
    const int mt    = blockIdx.x;      // M-tile

    // ---- zero LDS (covers K-pad cols and N-pad rows), then fill ----
    for (int i = tid; i < 64 * KPAD; i += 128) (&Wl[0][0])[i] = 0.0f;
    for (int i = tid; i < 16 * KPAD; i += 128) (&Xl[0][0])[i] = 0.0f;
    __syncthreads();
    for (int i = tid; i < G3 * IN_F; i += 128)
        Wl[i / IN_F][i % IN_F] = W[i];
    {
        const float* Xt = X + (size_t)mt * 16 * IN_F;
        for (int i = tid; i < 16 * IN_F; i += 128)
            Xl[i / IN_F][i % IN_F] = Xt[i];
    }
    __syncthreads();

    // ---- branch-free K loop ----
    v8f acc = {};
    const int ko   = khalf * 2;
    const int brow = wave * 16 + l16;
#pragma unroll
    for (int k0 = 0; k0 < KPAD; k0 += 4) {
        v2f a = *(const v2f*)&Xl[l16][k0 + ko];
        v2f b = *(const v2f*)&Wl[brow][k0 + ko];
        // 8 args: (neg_a, A, neg_b, B, c_mod, C, reuse_a, reuse_b)
        acc = __builtin_amdgcn_wmma_f32_16x16x4_f32(
            false, a, false, b, (short)0, acc, false, false);
    }

    // D layout: VGPR r -> M = r + 8*khalf, N = lane&15.
    // Transposed store: 8 consecutive M for fixed n -> contiguous b128 stores.
    const int n = wave * 16 + l16;
    if (n < G3) {
        const float bn = bias[n];
        const int mbase = mt * 16 + 8 * khalf;
        float* dst = gxT + (size_t)n * T_LEN + mbase;
#pragma unroll
        for (int r = 0; r < 8; ++r) dst[r] = acc[r] + bn;
    }
}

// ---------------------------------------------------------------------------
// Kernel 2: pipelined 4-layer GRU scan. One block, 4 waves; wave w = layer w,
// systolic skew t = tau - w. Hidden vectors flow through a double-buffered
// LDS ring; weights stay in registers (lane j holds gate rows j, j+20, j+40).
// Wave 3 computes the scalar readout from h[T-1].
// ---------------------------------------------------------------------------
__device__ __forceinline__ float sigm(float x) {
    return 1.0f / (1.0f + __expf(-x));
}
__device__ __forceinline__ float tanh_fast(float x) {
    const float e = __expf(-2.0f * x);
    return (1.0f - e) / (1.0f + e);
}

__global__ __launch_bounds__(128) void gru_scan(
    const float* __restrict__ gxT,     // [60, T] (transposed gate inputs)
    const float* __restrict__ w_hh0,   // [60, 20]
    const float* __restrict__ b_hh0,   // [60]
    const float* __restrict__ w_ih,    // [3, 60, 20]
    const float* __restrict__ w_hh,    // [3, 60, 20]
    const float* __restrict__ b_ih,    // [3, 60]
    const float* __restrict__ b_hh,    // [3, 60]
    const float* __restrict__ w_out,   // [20]
    const float* __restrict__ b_out,   // [1]
    float* __restrict__ out)           // [1]
{
    __shared__ float hbuf[NLAY][2][32];   // [layer][parity][h-vector]

    const int lane = threadIdx.x & 31;
    const int w    = threadIdx.x >> 5;    // layer index 0..3
    const bool act = lane < HID;
    const int  j   = act ? lane : 0;

    // ---- zero the ring buffers (h_{-1} = 0 for every layer) ----
    {
        float* hb = &hbuf[0][0][0];
        for (int i = threadIdx.x; i < NLAY * 2 * 32; i += 128) hb[i] = 0.0f;
    }

    // ---- load weights into registers ----
    float whr[HID], whz[HID], whn[HID];       // hidden->gate rows
    float uir[HID], uiz[HID], uin_[HID];      // input->gate rows (layers 1..3)
    float bhr, bhz, bhn, bir = 0.f, biz = 0.f, bin_ = 0.f;

    if (w == 0) {
#pragma unroll
        for (int k = 0; k < HID; ++k) {
            whr[k]  = w_hh0[(j          ) * HID + k];
            whz[k]  = w_hh0[(j + HID    ) * HID + k];
            whn[k]  = w_hh0[(j + 2 * HID) * HID + k];
            uir[k] = uiz[k] = uin_[k] = 0.0f;
        }
        bhr = b_hh0[j]; bhz = b_hh0[j + HID]; bhn = b_hh0[j + 2 * HID];
    } else {
        const float* whh = w_hh + (size_t)(w - 1) * G3 * HID;
        const float* wih = w_ih + (size_t)(w - 1) * G3 * HID;
#pragma unroll
        for (int k = 0; k < HID; ++k) {
            whr[k]  = whh[(j          ) * HID + k];
            whz[k]  = whh[(j + HID    ) * HID + k];
            whn[k]  = whh[(j + 2 * HID) * HID + k];
            uir[k]  = wih[(j          ) * HID + k];
            uiz[k]  = wih[(j + HID    ) * HID + k];
            uin_[k] = wih[(j + 2 * HID) * HID + k];
        }
        bhr = b_hh[(w - 1) * G3 + j];
        bhz = b_hh[(w - 1) * G3 + j + HID];
        bhn = b_hh[(w - 1) * G3 + j + 2 * HID];
        bir = b_ih[(w - 1) * G3 + j];
        biz = b_ih[(w - 1) * G3 + j + HID];
        bin_ = b_ih[(w - 1) * G3 + j + 2 * HID];
    }

    // Per-lane streaming pointers into the transposed gate-input array:
    // lane j walks three contiguous columns over t -> near-perfect locality.
    const float* gr = gxT + (size_t)(j          ) * T_LEN;
    const float* gz = gxT + (size_t)(j + HID    ) * T_LEN;
    const float* gn = gxT + (size_t)(j + 2 * HID) * T_LEN;

    float h = 0.0f;                       // this lane's hidden component

    // wave 0: prefetch gate inputs for t = 0
    float cxr = 0.f, cxz = 0.f, cxn = 0.f;
    if (w == 0) { cxr = gr[0]; cxz = gz[0]; cxn = gn[0]; }

    __syncthreads();

    for (int tau = 0; tau < T_LEN + NLAY - 1; ++tau) {
        const int t = tau - w;            // wave-uniform
        if (t >= 0 && t < T_LEN) {
            float xr, xz, xn;
            float nxr = 0.f, nxz = 0.f, nxn = 0.f;
            if (w == 0) {
                xr = cxr; xz = cxz; xn = cxn;
                // prefetch next timestep's gate inputs (hide HBM latency)
                if (t + 1 < T_LEN) {
                    nxr = gr[t + 1];
                    nxz = gz[t + 1];
                    nxn = gn[t + 1];
                }
            } else {
                // input projection of previous layer's h[t] (LDS broadcast)
                const float* hp = hbuf[w - 1][t & 1];
                xr = bir; xz = biz; xn = bin_;
#pragma unroll
                for (int k = 0; k < HID; ++k) {
                    const float hk = hp[k];
                    xr = fmaf(hk, uir[k],  xr);
                    xz = fmaf(hk, uiz[k],  xz);
                    xn = fmaf(hk, uin_[k], xn);
                }
            }

            // hidden projection of own h[t-1] (LDS broadcast, same ring)
            const float* hs = hbuf[w][(t - 1) & 1];
            float hr = bhr, hz = bhz, hn = bhn;
#pragma unroll
            for (int k = 0; k < HID; ++k) {
                const float hk = hs[k];
                hr = fmaf(hk, whr[k], hr);
                hz = fmaf(hk, whz[k], hz);
                hn = fmaf(hk, whn[k], hn);
            }

            const float r = sigm(xr + hr);
            const float z = sigm(xz + hz);
            const float n = tanh_fast(xn + r * hn);   // r gates (W_hn h + b_hn)
            h = (1.0f - z) * n + z * h;

            if (act) hbuf[w][t & 1][lane] = h;

            if (w == 0) { cxr = nxr; cxz = nxz; cxn = nxn; }
        }
        __syncthreads();                  // tick boundary (all 128 threads)
    }

    // readout: out = h3[T-1] . w_out + b_out
    if (w == NLAY - 1) {
        float v = act ? h * w_out[j] : 0.0f;
#pragma unroll
        for (int off = 16; off > 0; off >>= 1)
            v += __shfl_xor(v, off, 32);
        if (lane == 0) out[0] = v + b_out[0];
    }
}

// ---------------------------------------------------------------------------
// Host-side launcher
// ---------------------------------------------------------------------------
extern "C" void kernel_launch(void* const* d_in, const int* in_sizes, int n_in,
                              void* d_out, int out_size, void* d_ws, size_t ws_size,
                              hipStream_t stream) {
    const float* x     = (const float*)d_in[0];   // [1, T, 121]
    const float* w_ih0 = (const float*)d_in[1];   // [60, 121]
    const float* w_hh0 = (const float*)d_in[2];   // [60, 20]
    const float* b_ih0 = (const float*)d_in[3];   // [60]
    const float* b_hh0 = (const float*)d_in[4];   // [60]
    const float* w_ih  = (const float*)d_in[5];   // [3, 60, 20]
    const float* w_hh  = (const float*)d_in[6];   // [3, 60, 20]
    const float* b_ih  = (const float*)d_in[7];   // [3, 60]
    const float* b_hh  = (const float*)d_in[8];   // [3, 60]
    const float* w_out = (const float*)d_in[9];   // [1, 20]
    const float* b_out = (const float*)d_in[10];  // [1]

    float* gxT = (float*)d_ws;                    // [60, T] = 31.5 MB scratch

    // Phase 1: layer-0 input projection, WMMA fp32 GEMM (transposed output).
    gru_in_gemm<<<T_LEN / 16, 128, 0, stream>>>(x, w_ih0, b_ih0, gxT);

    // Phase 2: pipelined 4-layer recurrent scan + scalar readout.
    gru_scan<<<1, 128, 0, stream>>>(gxT, w_hh0, b_hh0, w_ih, w_hh,
                                    b_ih, b_hh, w_out, b_out, (float*)d_out);
}